// yolo_segment_target_48507360641097
// MI455X (gfx1250) — compile-verified
//
#include <hip/hip_runtime.h>

typedef __attribute__((ext_vector_type(2))) float v2f;
typedef __attribute__((ext_vector_type(4))) float v4f;
typedef __attribute__((ext_vector_type(8))) float v8f;

#define NROWS       8400
#define NCOLS       80
#define MASK_ELEMS  25600        // 160*160
#define CONF_THRESH 0.25f
#define SUM_BLOCKS  1024
#define SUM_THREADS 256
#define WAVE_CHUNK  512          // floats consumed per wave per iteration (4 x b128 per lane)

// ---------------- init: first_invalid = NROWS ----------------
__global__ void init_ws_kernel(int* __restrict__ first_invalid, int L) {
    if (blockIdx.x == 0 && threadIdx.x == 0) first_invalid[0] = L;
}

// ---------------- pass 1: per-row max, atomicMin of first failing row ----------------
__global__ void rowmax_kernel(const float* __restrict__ post,
                              int* __restrict__ first_invalid) {
    int gtid = blockIdx.x * blockDim.x + threadIdx.x;
    int wave = gtid >> 5;
    int lane = threadIdx.x & 31;
    if (wave >= NROWS) return;
    const float* row = post + (size_t)wave * NCOLS;
    float m = -1e30f;
    for (int c = lane; c < NCOLS; c += 32) m = fmaxf(m, row[c]);
    // wave32 shuffle reduction
    for (int off = 16; off > 0; off >>= 1)
        m = fmaxf(m, __shfl_xor(m, off, 32));
    if (lane == 0 && m < CONF_THRESH) atomicMin(first_invalid, wave);
}

// ---------------- pass 2: stream masks, accumulate with V_WMMA_F32_16X16X4_F32 ----------------
// B = ones(4x16)  =>  D = A*B + C accumulates row-sums of A; summing the whole
// 16x16 accumulator gives 16x the element sum (each A element hit once per N column).
__global__ void masksum_kernel(const float* __restrict__ mask,
                               const int* __restrict__ first_invalid,
                               float* __restrict__ partials) {
    __shared__ float wsum[SUM_THREADS / 32];

    const int L = first_invalid[0];                       // uniform scalar
    const long long total = (long long)L * MASK_ELEMS;    // multiple of WAVE_CHUNK

    const int lane  = threadIdx.x & 31;
    const long long wave  = (long long)(blockIdx.x * blockDim.x + threadIdx.x) >> 5;
    const long long nwave = ((long long)gridDim.x * blockDim.x) >> 5;
    const long long stride = nwave * WAVE_CHUNK;

    v8f acc = {0.f, 0.f, 0.f, 0.f, 0.f, 0.f, 0.f, 0.f};
    const v2f ones = {1.0f, 1.0f};

    for (long long base = wave * WAVE_CHUNK; base < total; base += stride) {
        const float* p = mask + base + (size_t)lane * 4;
        // 4 non-temporal b128 loads per lane: 512 floats per wave iteration
        v4f d0 = __builtin_nontemporal_load((const v4f*)(p + 0 * 128));
        v4f d1 = __builtin_nontemporal_load((const v4f*)(p + 1 * 128));
        v4f d2 = __builtin_nontemporal_load((const v4f*)(p + 2 * 128));
        v4f d3 = __builtin_nontemporal_load((const v4f*)(p + 3 * 128));

        v2f a0 = {d0[0], d0[1]}, a1 = {d0[2], d0[3]};
        v2f a2 = {d1[0], d1[1]}, a3 = {d1[2], d1[3]};
        v2f a4 = {d2[0], d2[1]}, a5 = {d2[2], d2[3]};
        v2f a6 = {d3[0], d3[1]}, a7 = {d3[2], d3[3]};

        acc = __builtin_amdgcn_wmma_f32_16x16x4_f32(false, a0, false, ones, (short)0, acc, false, false);
        acc = __builtin_amdgcn_wmma_f32_16x16x4_f32(false, a1, false, ones, (short)0, acc, false, false);
        acc = __builtin_amdgcn_wmma_f32_16x16x4_f32(false, a2, false, ones, (short)0, acc, false, false);
        acc = __builtin_amdgcn_wmma_f32_16x16x4_f32(false, a3, false, ones, (short)0, acc, false, false);
        acc = __builtin_amdgcn_wmma_f32_16x16x4_f32(false, a4, false, ones, (short)0, acc, false, false);
        acc = __builtin_amdgcn_wmma_f32_16x16x4_f32(false, a5, false, ones, (short)0, acc, false, false);
        acc = __builtin_amdgcn_wmma_f32_16x16x4_f32(false, a6, false, ones, (short)0, acc, false, false);
        acc = __builtin_amdgcn_wmma_f32_16x16x4_f32(false, a7, false, ones, (short)0, acc, false, false);
    }

    // per-lane sum of the 8 accumulator VGPRs
    float s = 0.f;
    #pragma unroll
    for (int i = 0; i < 8; ++i) s += acc[i];
    // wave32 reduction (fixed order -> deterministic)
    for (int off = 16; off > 0; off >>= 1)
        s += __shfl_xor(s, off, 32);
    if (lane == 0) wsum[threadIdx.x >> 5] = s;
    __syncthreads();
    if (threadIdx.x == 0) {
        float b = 0.f;
        #pragma unroll
        for (int w = 0; w < SUM_THREADS / 32; ++w) b += wsum[w];
        partials[blockIdx.x] = b;
    }
}

// ---------------- pass 3: deterministic final reduction + scale ----------------
__global__ void final_reduce_kernel(const float* __restrict__ partials,
                                    float* __restrict__ out) {
    __shared__ float sm[SUM_THREADS];
    float s = 0.f;
    for (int i = threadIdx.x; i < SUM_BLOCKS; i += SUM_THREADS) s += partials[i];
    sm[threadIdx.x] = s;
    __syncthreads();
    for (int off = SUM_THREADS / 2; off > 0; off >>= 1) {
        if (threadIdx.x < off) sm[threadIdx.x] += sm[threadIdx.x + off];
        __syncthreads();
    }
    if (threadIdx.x == 0)
        out[0] = sm[0] * (1.0f / (16.0f * (float)MASK_ELEMS));
}

extern "C" void kernel_launch(void* const* d_in, const int* in_sizes, int n_in,
                              void* d_out, int out_size, void* d_ws, size_t ws_size,
                              hipStream_t stream) {
    const float* post_result   = (const float*)d_in[0];  // [8400, 80]
    const float* pre_post_mask = (const float*)d_in[2];  // [8400, 160, 160]
    float* out = (float*)d_out;                           // scalar

    // workspace layout: [0]   int   first_invalid
    //                   [256] float partials[SUM_BLOCKS]
    int*   ws_first    = (int*)d_ws;
    float* ws_partials = (float*)((char*)d_ws + 256);

    (void)in_sizes; (void)n_in; (void)out_size; (void)ws_size;

    init_ws_kernel<<<1, 1, 0, stream>>>(ws_first, NROWS);

    int rowmax_blocks = (NROWS * 32 + 255) / 256;  // one wave per row
    rowmax_kernel<<<rowmax_blocks, 256, 0, stream>>>(post_result, ws_first);

    masksum_kernel<<<SUM_BLOCKS, SUM_THREADS, 0, stream>>>(pre_post_mask, ws_first, ws_partials);

    final_reduce_kernel<<<1, SUM_THREADS, 0, stream>>>(ws_partials, out);
}